// DGCNN_2345052143618
// MI455X (gfx1250) — compile-verified
//
#include <hip/hip_runtime.h>
#include <cstdint>
#include <cstddef>

#define NPTS  4096
#define BATCH 4
#define KNN   64
#define KROWS 3   // rows per KNN block (3*16KB dist in LDS)

typedef float v2f __attribute__((ext_vector_type(2)));
typedef float v8f __attribute__((ext_vector_type(8)));

__device__ __forceinline__ void lds_fence() {
  asm volatile("s_wait_dscnt 0" ::: "memory");
}

// ---------------------------------------------------------------- sq norms
template<int C>
__global__ void sq_kernel(const float* __restrict__ xt, float* __restrict__ sq) {
  int i = blockIdx.x * blockDim.x + threadIdx.x;  // over B*N
  if (i >= BATCH * NPTS) return;
  const float* row = xt + (size_t)i * C;
  float s = 0.f;
#pragma unroll
  for (int c = 0; c < C; ++c) { float v = row[c]; s += v * v; }
  sq[i] = s;
}

// ------------------------------------------- transpose XC rows -> XT[N][64]
__global__ void transpose64_kernel(const float* __restrict__ xc, int rowbase,
                                   float* __restrict__ xt) {
  int n = blockIdx.x * blockDim.x + threadIdx.x;
  int b = blockIdx.y;
  if (n >= NPTS) return;
  const float* src = xc + (size_t)b * 320 * NPTS + (size_t)rowbase * NPTS + n;
  float* dst = xt + ((size_t)b * NPTS + n) * 64;
#pragma unroll 8
  for (int c = 0; c < 64; ++c) dst[c] = src[(size_t)c * NPTS];
}

// ---------------------------------------------------------------- KNN
// block = 128 threads (4 waves); KROWS rows per block; dist kept in LDS.
template<int C>
__global__ void knn_kernel(const float* __restrict__ xt, const float* __restrict__ sq,
                           int* __restrict__ idx) {
  extern __shared__ char smem[];
  float* dist = (float*)smem;                     // [KROWS][NPTS]
  float* ctr  = dist + KROWS * NPTS;              // [KROWS][C]
  float* sqr  = ctr + KROWS * C;                  // [KROWS]
  const int tid = threadIdx.x;
  const int b = blockIdx.y;
  const int r0 = blockIdx.x * KROWS;
  const float* xb  = xt + (size_t)b * NPTS * C;
  const float* sqb = sq + (size_t)b * NPTS;

  for (int e = tid; e < KROWS * C; e += 128) {
    int r = e / C, c = e % C;
    ctr[e] = (r0 + r < NPTS) ? xb[(size_t)(r0 + r) * C + c] : 0.f;
  }
  if (tid < KROWS) sqr[tid] = (r0 + tid < NPTS) ? sqb[r0 + tid] : 0.f;
  __syncthreads();

  for (int m = tid; m < NPTS; m += 128) {
    if (m + 128 < NPTS)  // warm next feature row (global_prefetch_b8)
      __builtin_prefetch(xb + (size_t)(m + 128) * C, 0, 3);
    float acc[KROWS];
#pragma unroll
    for (int r = 0; r < KROWS; ++r) acc[r] = 0.f;
    const float* xm = xb + (size_t)m * C;
    if constexpr (C % 4 == 0) {
      for (int c = 0; c < C; c += 4) {
        float4 xv = *(const float4*)(xm + c);
#pragma unroll
        for (int r = 0; r < KROWS; ++r) {
          float4 cv = *(const float4*)(ctr + r * C + c);
          acc[r] += xv.x * cv.x + xv.y * cv.y + xv.z * cv.z + xv.w * cv.w;
        }
      }
    } else {
#pragma unroll
      for (int c = 0; c < C; ++c) {
        float xv = xm[c];
#pragma unroll
        for (int r = 0; r < KROWS; ++r) acc[r] += xv * ctr[r * C + c];
      }
    }
    float sm = sqb[m];
#pragma unroll
    for (int r = 0; r < KROWS; ++r)
      dist[r * NPTS + m] = sqr[r] - 2.f * acc[r] + sm;
  }
  __syncthreads();

  const int wave = tid >> 5, lane = tid & 31;
  if (wave < KROWS && (r0 + wave) < NPTS) {
    float* drow = dist + wave * NPTS;
    int* orow = idx + ((size_t)b * NPTS + (r0 + wave)) * KNN;
    for (int round = 0; round < KNN; ++round) {
      float lm = 3.4e38f; int li = NPTS;
      // vectorized scan: ds_load_b128, indices ascending within lane
      for (int j = 0; j < NPTS / 128; ++j) {
        int mb = j * 128 + lane * 4;
        float4 v4 = *(const float4*)(drow + mb);
        if (v4.x < lm) { lm = v4.x; li = mb; }
        if (v4.y < lm) { lm = v4.y; li = mb + 1; }
        if (v4.z < lm) { lm = v4.z; li = mb + 2; }
        if (v4.w < lm) { lm = v4.w; li = mb + 3; }
      }
#pragma unroll
      for (int off = 16; off >= 1; off >>= 1) {
        float ov = __shfl_xor(lm, off, 32);
        int   oi = __shfl_xor(li, off, 32);
        if (ov < lm || (ov == lm && oi < li)) { lm = ov; li = oi; }
      }
      if (lane == 0) { orow[round] = li; drow[li] = 3.4e38f; }
      lds_fence();
    }
  }
}

// ---------------------------------------------------------------- edge conv
// y = W1 @ nbr + (W2-W1) @ ctr ; out = max_k lrelu(bn(y))
// block = 256 (8 waves): 16 output channels x 16 points; wave owns 2 points.
template<int C, int O>
__global__ void edgeconv_kernel(const float* __restrict__ xt, const int* __restrict__ idx,
                                const float* __restrict__ W, const float* __restrict__ g,
                                const float* __restrict__ bb, float* __restrict__ xc,
                                int rowbase) {
  constexpr int CP = (C + 3) & ~3;
  __shared__ float w1l[16][CP];
  __shared__ float wdl[16][CP];
  __shared__ float ctr[16][CP];
  __shared__ float nbr[8][16][CP];
  __shared__ float biasl[16][16];
  __shared__ int   idxl[16][KNN];
  __shared__ float gsv[16], bbv[16];
  const int tid = threadIdx.x;
  const int b  = blockIdx.z;
  const int n0 = blockIdx.y * 16;
  const int ob = blockIdx.x * 16;
  const float* xb = xt + (size_t)b * NPTS * C;
  const float inv = rsqrtf(1.0f + 1e-5f);

  for (int e = tid; e < 16 * CP; e += 256) {
    int m = e / CP, c = e % CP;
    float a = 0.f, d = 0.f;
    if (c < C) {
      float w1v = W[(size_t)(ob + m) * (2 * C) + c];
      float w2v = W[(size_t)(ob + m) * (2 * C) + C + c];
      a = w1v; d = w2v - w1v;
    }
    w1l[m][c] = a; wdl[m][c] = d;
  }
  for (int e = tid; e < 16 * CP; e += 256) {
    int p = e / CP, c = e % CP;
    ctr[p][c] = (c < C) ? xb[(size_t)(n0 + p) * C + c] : 0.f;
  }
  for (int e = tid; e < 16 * KNN; e += 256) {
    int p = e / KNN, k = e % KNN;
    int nv = idx[((size_t)b * NPTS + n0 + p) * KNN + k];
    idxl[p][k] = nv;
  }
  if (tid < 16) { gsv[tid] = g[ob + tid] * inv; bbv[tid] = bb[ob + tid]; }
  __syncthreads();

  const int wave = tid >> 5, lane = tid & 31;
  const int mrow = lane & 15;
  const int hi = lane >> 4, hi2 = hi * 2;

  if (wave == 0) {  // bias tile = (W2-W1) @ ctr  (16x16)
    v8f acc = {};
#pragma unroll
    for (int c0 = 0; c0 < CP; c0 += 4) {
      v2f a  = *(const v2f*)&wdl[mrow][c0 + hi2];
      v2f bf = *(const v2f*)&ctr[mrow][c0 + hi2];
      acc = __builtin_amdgcn_wmma_f32_16x16x4_f32(false, a, false, bf, (short)0, acc,
                                                  false, false);
    }
#pragma unroll
    for (int i = 0; i < 8; ++i) biasl[i + 8 * hi][mrow] = acc[i];
  }
  __syncthreads();

  for (int pp = 0; pp < 2; ++pp) {
    const int p = wave * 2 + pp;
    const int n = n0 + p;
    // warm all 64 neighbor rows for this point (global_prefetch_b8)
    for (int e = lane; e < KNN; e += 32)
      __builtin_prefetch(xb + (size_t)idxl[p][e] * C, 0, 3);

    float rmax[8];
#pragma unroll
    for (int i = 0; i < 8; ++i) rmax[i] = -3.4e38f;

    for (int q = 0; q < 4; ++q) {  // 4 chunks of 16 neighbors
      if constexpr (C % 4 == 0) {
        for (int e = lane; e < 16 * (CP / 4); e += 32) {
          int kp = e / (CP / 4), c4 = e % (CP / 4);
          int nb = idxl[p][q * 16 + kp];
          *(float4*)&nbr[wave][kp][c4 * 4] = *(const float4*)(xb + (size_t)nb * C + c4 * 4);
        }
      } else {
        for (int e = lane; e < 16 * CP; e += 32) {
          int kp = e / CP, c = e % CP;
          int nb = idxl[p][q * 16 + kp];
          nbr[wave][kp][c] = (c < C) ? xb[(size_t)nb * C + c] : 0.f;
        }
      }
      lds_fence();  // intra-wave LDS RAW before fragment reads

      v8f acc = {};
#pragma unroll
      for (int c0 = 0; c0 < CP; c0 += 4) {
        v2f a  = *(const v2f*)&w1l[mrow][c0 + hi2];
        v2f bf = *(const v2f*)&nbr[wave][mrow][c0 + hi2];
        acc = __builtin_amdgcn_wmma_f32_16x16x4_f32(false, a, false, bf, (short)0, acc,
                                                    false, false);
      }
#pragma unroll
      for (int i = 0; i < 8; ++i) {
        int o_r = i + 8 * hi;
        float z = (acc[i] + biasl[o_r][p]) * gsv[o_r] + bbv[o_r];
        float act = fmaxf(z, 0.2f * z);
        rmax[i] = fmaxf(rmax[i], act);
      }
    }
#pragma unroll
    for (int i = 0; i < 8; ++i) {
      float v = rmax[i];
      v = fmaxf(v, __shfl_xor(v, 1, 32));
      v = fmaxf(v, __shfl_xor(v, 2, 32));
      v = fmaxf(v, __shfl_xor(v, 4, 32));
      v = fmaxf(v, __shfl_xor(v, 8, 32));
      rmax[i] = v;
    }
    if (mrow == 0) {
      float* dst = xc + (size_t)b * 320 * NPTS + (size_t)(rowbase + ob) * NPTS + n;
#pragma unroll
      for (int i = 0; i < 8; ++i) dst[(size_t)(i + 8 * hi) * NPTS] = rmax[i];
    }
  }
}

// ---------------------------------------------- 1024x320 GEMM + fused pooling
__global__ void gemm5_pool_kernel(const float* __restrict__ xc, const float* __restrict__ w5,
                                  const float* __restrict__ g5, const float* __restrict__ b5,
                                  float* __restrict__ pmax, float* __restrict__ psum) {
  __shared__ float w5t[16][320];
  __shared__ float gsv[16], bbv[16];
  const int tid = threadIdx.x;  // 128
  const int b  = blockIdx.z;
  const int ob = blockIdx.y * 16;
  const int nb = blockIdx.x;    // 64 blocks of 64 cols
  const float inv = rsqrtf(1.0f + 1e-5f);
  const float* xb = xc + (size_t)b * 320 * NPTS;
  // warm the 320 row segments this block will stream (global_prefetch_b8)
  for (int r = tid; r < 320; r += 128)
    __builtin_prefetch(xb + (size_t)r * NPTS + nb * 64, 0, 3);
  for (int e = tid; e < 16 * 320; e += 128) {
    int m = e / 320, c = e % 320;
    w5t[m][c] = w5[(size_t)(ob + m) * 320 + c];
  }
  if (tid < 16) { gsv[tid] = g5[ob + tid] * inv; bbv[tid] = b5[ob + tid]; }
  __syncthreads();

  const int wave = tid >> 5, lane = tid & 31;
  const int mrow = lane & 15, hi = lane >> 4, hi2 = hi * 2;
  const int ncol = nb * 64 + wave * 16 + mrow;
  v8f acc = {};
  for (int c0 = 0; c0 < 320; c0 += 4) {
    v2f a = *(const v2f*)&w5t[mrow][c0 + hi2];
    v2f bf;
    bf[0] = xb[(size_t)(c0 + hi2) * NPTS + ncol];
    bf[1] = xb[(size_t)(c0 + hi2 + 1) * NPTS + ncol];
    acc = __builtin_amdgcn_wmma_f32_16x16x4_f32(false, a, false, bf, (short)0, acc,
                                                false, false);
  }
  float amax[8], asum[8];
#pragma unroll
  for (int i = 0; i < 8; ++i) {
    int o_r = i + 8 * hi;
    float z = acc[i] * gsv[o_r] + bbv[o_r];
    float act = fmaxf(z, 0.2f * z);
    amax[i] = act; asum[i] = act;
  }
#pragma unroll
  for (int i = 0; i < 8; ++i) {
#pragma unroll
    for (int off = 1; off <= 8; off <<= 1) {
      amax[i] = fmaxf(amax[i], __shfl_xor(amax[i], off, 32));
      asum[i] += __shfl_xor(asum[i], off, 32);
    }
  }
  if (mrow == 0) {
    int nb4 = nb * 4 + wave;  // 0..255 partial slot
#pragma unroll
    for (int i = 0; i < 8; ++i) {
      int o = ob + i + 8 * hi;
      pmax[((size_t)b * 1024 + o) * 256 + nb4] = amax[i];
      psum[((size_t)b * 1024 + o) * 256 + nb4] = asum[i];
    }
  }
}

__global__ void pool_reduce_kernel(const float* __restrict__ pmax, const float* __restrict__ psum,
                                   float* __restrict__ gmax, float* __restrict__ gavg) {
  int i = blockIdx.x * blockDim.x + threadIdx.x;  // B*1024
  if (i >= BATCH * 1024) return;
  const float* pm = pmax + (size_t)i * 256;
  const float* ps = psum + (size_t)i * 256;
  float m = -3.4e38f, s = 0.f;
  for (int j = 0; j < 256; ++j) { m = fmaxf(m, pm[j]); s += ps[j]; }
  gmax[i] = m; gavg[i] = s * (1.0f / NPTS);
}

// ---------------------------------------------------------------- MLP head
__global__ void head_kernel(const float* __restrict__ gmax, const float* __restrict__ gavg,
                            const float* __restrict__ wl1, const float* __restrict__ g6,
                            const float* __restrict__ b6, const float* __restrict__ wl2,
                            const float* __restrict__ bl2, const float* __restrict__ g7,
                            const float* __restrict__ b7, const float* __restrict__ wl3,
                            const float* __restrict__ bl3, float* __restrict__ out) {
  __shared__ float h[2048];
  __shared__ float h1[512];
  __shared__ float h2[256];
  __shared__ float o5[5];
  const int tid = threadIdx.x;  // 256
  const int b = blockIdx.x;
  const float inv = rsqrtf(1.0f + 1e-5f);
  for (int c = tid; c < 1024; c += 256) {
    h[c]        = gmax[b * 1024 + c];
    h[1024 + c] = gavg[b * 1024 + c];
  }
  __syncthreads();
  for (int o = tid; o < 512; o += 256) {
    const float* wr = wl1 + (size_t)o * 2048;
    float acc = 0.f;
    for (int c = 0; c < 2048; ++c) acc += h[c] * wr[c];
    float z = acc * (g6[o] * inv) + b6[o];
    h1[o] = fmaxf(z, 0.2f * z);
  }
  __syncthreads();
  {
    int o = tid;
    const float* wr = wl2 + (size_t)o * 512;
    float acc = 0.f;
    for (int c = 0; c < 512; ++c) acc += h1[c] * wr[c];
    acc += bl2[o];
    float z = acc * (g7[o] * inv) + b7[o];
    h2[o] = fmaxf(z, 0.2f * z);
  }
  __syncthreads();
  if (tid < 5) {
    const float* wr = wl3 + (size_t)tid * 256;
    float acc = 0.f;
    for (int c = 0; c < 256; ++c) acc += h2[c] * wr[c];
    o5[tid] = acc + bl3[tid];
  }
  __syncthreads();
  float* ob = out + (size_t)b * NPTS * 5;
  for (int i = tid; i < NPTS * 5; i += 256) ob[i] = o5[i % 5];
}

// ---------------------------------------------------------------- launch
extern "C" void kernel_launch(void* const* d_in, const int* in_sizes, int n_in,
                              void* d_out, int out_size, void* d_ws, size_t ws_size,
                              hipStream_t stream) {
  (void)in_sizes; (void)n_in; (void)out_size; (void)ws_size;
  const float* xyz = (const float*)d_in[0];
  const float* w1  = (const float*)d_in[1];
  const float* w2  = (const float*)d_in[2];
  const float* w3  = (const float*)d_in[3];
  const float* w4  = (const float*)d_in[4];
  const float* w5  = (const float*)d_in[5];
  const float* g1 = (const float*)d_in[6],  *b1 = (const float*)d_in[7];
  const float* g2 = (const float*)d_in[8],  *b2 = (const float*)d_in[9];
  const float* g3 = (const float*)d_in[10], *b3 = (const float*)d_in[11];
  const float* g4 = (const float*)d_in[12], *b4 = (const float*)d_in[13];
  const float* g5 = (const float*)d_in[14], *b5 = (const float*)d_in[15];
  const float* g6 = (const float*)d_in[16], *b6 = (const float*)d_in[17];
  const float* g7 = (const float*)d_in[18], *b7 = (const float*)d_in[19];
  const float* wl1 = (const float*)d_in[20];
  const float* wl2 = (const float*)d_in[21];
  const float* bl2 = (const float*)d_in[22];
  const float* wl3 = (const float*)d_in[23];
  const float* bl3 = (const float*)d_in[24];
  float* out = (float*)d_out;

  char* wsB = (char*)d_ws;
  size_t off = 0;
  auto carve = [&](size_t bytes) -> void* {
    void* p = wsB + off;
    off = (off + bytes + 255) & ~(size_t)255;
    return p;
  };
  float* XT   = (float*)carve(sizeof(float) * BATCH * NPTS * 64);
  float* XC   = (float*)carve(sizeof(float) * BATCH * 320 * NPTS);
  float* SQ   = (float*)carve(sizeof(float) * BATCH * NPTS);
  int*   IDX  = (int*)  carve(sizeof(int)   * BATCH * NPTS * KNN);
  float* PMAX = (float*)carve(sizeof(float) * BATCH * 1024 * 256);
  float* PSUM = (float*)carve(sizeof(float) * BATCH * 1024 * 256);
  float* GMAX = (float*)carve(sizeof(float) * BATCH * 1024);
  float* GAVG = (float*)carve(sizeof(float) * BATCH * 1024);

  const dim3 knnGrid((NPTS + KROWS - 1) / KROWS, BATCH);
  const size_t knn3Smem  = (size_t)KROWS * NPTS * 4 + (size_t)KROWS * 3 * 4 + KROWS * 4;
  const size_t knn64Smem = (size_t)KROWS * NPTS * 4 + (size_t)KROWS * 64 * 4 + KROWS * 4;

  // ---- layer 1 (xyz as XT, C=3)
  sq_kernel<3><<<(BATCH * NPTS + 255) / 256, 256, 0, stream>>>(xyz, SQ);
  knn_kernel<3><<<knnGrid, 128, knn3Smem, stream>>>(xyz, SQ, IDX);
  edgeconv_kernel<3, 64><<<dim3(4, NPTS / 16, BATCH), 256, 0, stream>>>(
      xyz, IDX, w1, g1, b1, XC, 0);
  // ---- layer 2 (input rows 0..63)
  transpose64_kernel<<<dim3(NPTS / 256, BATCH), 256, 0, stream>>>(XC, 0, XT);
  sq_kernel<64><<<(BATCH * NPTS + 255) / 256, 256, 0, stream>>>(XT, SQ);
  knn_kernel<64><<<knnGrid, 128, knn64Smem, stream>>>(XT, SQ, IDX);
  edgeconv_kernel<64, 64><<<dim3(4, NPTS / 16, BATCH), 256, 0, stream>>>(
      XT, IDX, w2, g2, b2, XC, 64);
  // ---- layer 3 (input rows 64..127)
  transpose64_kernel<<<dim3(NPTS / 256, BATCH), 256, 0, stream>>>(XC, 64, XT);
  sq_kernel<64><<<(BATCH * NPTS + 255) / 256, 256, 0, stream>>>(XT, SQ);
  knn_kernel<64><<<knnGrid, 128, knn64Smem, stream>>>(XT, SQ, IDX);
  edgeconv_kernel<64, 64><<<dim3(4, NPTS / 16, BATCH), 256, 0, stream>>>(
      XT, IDX, w3, g3, b3, XC, 128);
  // ---- layer 4 (input rows 128..191, O=128)
  transpose64_kernel<<<dim3(NPTS / 256, BATCH), 256, 0, stream>>>(XC, 128, XT);
  sq_kernel<64><<<(BATCH * NPTS + 255) / 256, 256, 0, stream>>>(XT, SQ);
  knn_kernel<64><<<knnGrid, 128, knn64Smem, stream>>>(XT, SQ, IDX);
  edgeconv_kernel<64, 128><<<dim3(8, NPTS / 16, BATCH), 256, 0, stream>>>(
      XT, IDX, w4, g4, b4, XC, 192);
  // ---- conv5 GEMM + pooling
  gemm5_pool_kernel<<<dim3(64, 64, BATCH), 128, 0, stream>>>(XC, w5, g5, b5, PMAX, PSUM);
  pool_reduce_kernel<<<(BATCH * 1024 + 255) / 256, 256, 0, stream>>>(PMAX, PSUM, GMAX, GAVG);
  // ---- MLP head + broadcast
  head_kernel<<<BATCH, 256, 0, stream>>>(GMAX, GAVG, wl1, g6, b6, wl2, bl2, g7, b7,
                                         wl3, bl3, out);
}